// RBERT_24257975288387
// MI455X (gfx1250) — compile-verified
//
#include <hip/hip_runtime.h>
#include <hip/hip_bf16.h>
#include <cmath>

typedef __attribute__((ext_vector_type(2))) float v2f;
typedef __attribute__((ext_vector_type(8))) float v8f;

#define B_SZ   2048
#define S_SZ   256
#define D_SZ   768
#define F_SZ   (3 * D_SZ)   // 2304
#define H_SZ   100
#define NTILES 7            // 7*16 = 112 >= 100

// ---------------------------------------------------------------------------
// Kernel 1: span means + tanh + feature row construction.
// One block per batch row; 256 threads each own 3 of 768 channels.
// ---------------------------------------------------------------------------
__global__ __launch_bounds__(256) void span_feat_kernel(
    const float* __restrict__ hidden,   // (B, S, D)
    const int*   __restrict__ locs,     // (B, 4) int32
    float*       __restrict__ feat)     // (B, 3D) workspace
{
    const int b = blockIdx.x;
    const int t = threadIdx.x;
    const float* hb = hidden + (size_t)b * S_SZ * D_SZ;

    const int l0 = locs[b * 4 + 0];
    const int l1 = locs[b * 4 + 1];
    const int l2 = locs[b * 4 + 2];
    const int l3 = locs[b * 4 + 3];
    // mask is (pos > lo) & (pos < hi)  ->  pos in [lo+1, hi)
    const int lo1 = l0 + 1, hi1 = l1;
    const int lo2 = l2 + 1, hi2 = l3;
    const float c1 = fmaxf((float)(hi1 - lo1), 1.0f);
    const float c2 = fmaxf((float)(hi2 - lo2), 1.0f);

    float s1[3] = {0.f, 0.f, 0.f};
    float s2[3] = {0.f, 0.f, 0.f};

    for (int p = lo1; p < hi1; ++p) {
        const float* row = hb + (size_t)p * D_SZ;
        __builtin_prefetch(row + D_SZ, 0, 1);   // global_prefetch_b8: next row
        #pragma unroll
        for (int j = 0; j < 3; ++j) s1[j] += row[t + 256 * j];
    }
    for (int p = lo2; p < hi2; ++p) {
        const float* row = hb + (size_t)p * D_SZ;
        __builtin_prefetch(row + D_SZ, 0, 1);
        #pragma unroll
        for (int j = 0; j < 3; ++j) s2[j] += row[t + 256 * j];
    }

    float* fr = feat + (size_t)b * F_SZ;
    #pragma unroll
    for (int j = 0; j < 3; ++j) {
        const int d = t + 256 * j;
        const float e1 = tanhf(s1[j] / c1);
        const float e2 = tanhf(s2[j] / c2);
        fr[d]              = hb[d];        // sent_emb = hidden[b, 0, :]
        fr[D_SZ + d]       = e1 - e2;      // diff
        fr[2 * D_SZ + d]   = e1 * e2;      // prod
    }
}

// ---------------------------------------------------------------------------
// Kernel 2: h = feat @ W1 + b1 via V_WMMA_F32_16X16X4_F32, then out = h@W2+b2.
// One block per 16 batch rows; 7 waves, wave w owns N-tile [16w, 16w+16).
// A-layout (16x4 f32): lane<16 -> M=lane, K=k+0/k+1; lane>=16 -> M=lane-16, K=k+2/k+3.
// B-layout (4x16 f32): mirrored: v0/v1 hold rows K=k+khalf, k+khalf+1 at N=lane&15.
// C-layout (16x16 f32): VGPR v: lanes 0-15 -> M=v, lanes 16-31 -> M=v+8; N=n0+(lane&15).
// Padding (cols 100..111) handled branchlessly: loads use a clamped in-bounds
// address, then v_cndmask selects 0 -> no exec-mask branching in the hot loop.
// ---------------------------------------------------------------------------
__global__ __launch_bounds__(32 * NTILES) void gemm_head_kernel(
    const float* __restrict__ feat,  // (B, 2304)
    const float* __restrict__ W1,    // (2304, 100) row-major
    const float* __restrict__ b1,    // (100)
    const float* __restrict__ W2,    // (100, 1)
    const float* __restrict__ b2,    // (1)
    float*       __restrict__ out)   // (B, 1)
{
    __shared__ float hTile[16][NTILES * 16];   // 16 x 112

    const int m0   = blockIdx.x * 16;
    const int tid  = threadIdx.x;
    const int wave = tid >> 5;          // 0..6 -> N tile
    const int lane = tid & 31;
    const int nl   = lane & 15;         // N (and A's M) within tile
    const int n0   = wave * 16;
    const int n    = n0 + nl;
    const int khalf = (lane >> 4) * 2;  // 0 or 2
    const bool nvalid = (n < H_SZ);
    const int  ncl    = nvalid ? n : 0; // clamped, always in-bounds

    const float* aptr = feat + (size_t)(m0 + nl) * F_SZ + khalf;
    const float* bptr = W1 + (size_t)khalf * H_SZ + ncl;

    v8f c0 = {0.f, 0.f, 0.f, 0.f, 0.f, 0.f, 0.f, 0.f};
    v8f c1 = c0;

    #pragma unroll 4
    for (int k = 0; k < F_SZ; k += 8) {
        // Two independent K-steps per iteration -> two accumulator chains.
        v2f a0 = *(const v2f*)(aptr + k);           // 8B-aligned global_load_b64
        v2f a1 = *(const v2f*)(aptr + k + 4);
        const float w00 = bptr[(size_t)(k    ) * H_SZ];
        const float w01 = bptr[(size_t)(k + 1) * H_SZ];
        const float w10 = bptr[(size_t)(k + 4) * H_SZ];
        const float w11 = bptr[(size_t)(k + 5) * H_SZ];
        v2f bm0, bm1;
        bm0.x = nvalid ? w00 : 0.0f;    // v_cndmask, no branch
        bm0.y = nvalid ? w01 : 0.0f;
        bm1.x = nvalid ? w10 : 0.0f;
        bm1.y = nvalid ? w11 : 0.0f;
        // (neg_a, A, neg_b, B, c_mod, C, reuse_a, reuse_b)
        c0 = __builtin_amdgcn_wmma_f32_16x16x4_f32(
                 false, a0, false, bm0, (short)0, c0, false, false);
        c1 = __builtin_amdgcn_wmma_f32_16x16x4_f32(
                 false, a1, false, bm1, (short)0, c1, false, false);
    }

    const float bias  = nvalid ? b1[ncl] : 0.0f;
    const int   mbase = (lane >> 4) * 8;
    #pragma unroll
    for (int v = 0; v < 8; ++v)
        hTile[mbase + v][n] = c0[v] + c1[v] + bias;

    __syncthreads();

    // Final head: out[m0+m] = sum_n h[m,n] * W2[n] + b2
    if (tid < 16) {
        float s = b2[0];
        #pragma unroll 4
        for (int j = 0; j < H_SZ; ++j)
            s += hTile[tid][j] * W2[j];
        out[m0 + tid] = s;
    }
}

// ---------------------------------------------------------------------------
extern "C" void kernel_launch(void* const* d_in, const int* in_sizes, int n_in,
                              void* d_out, int out_size, void* d_ws, size_t ws_size,
                              hipStream_t stream) {
    const float* hidden = (const float*)d_in[0];  // (B,S,D) f32
    const int*   locs   = (const int*)  d_in[1];  // (B,4) int32 (JAX x64 off)
    const float* W1     = (const float*)d_in[2];  // (2304,100)
    const float* b1     = (const float*)d_in[3];  // (100)
    const float* W2     = (const float*)d_in[4];  // (100,1)
    const float* b2     = (const float*)d_in[5];  // (1)
    float*       out    = (float*)d_out;          // (B,1)
    float*       feat   = (float*)d_ws;           // (B,2304) = ~18.9 MB scratch

    span_feat_kernel<<<B_SZ, 256, 0, stream>>>(hidden, locs, feat);
    gemm_head_kernel<<<B_SZ / 16, 32 * NTILES, 0, stream>>>(feat, W1, b1, W2, b2, out);
}